// MVDR_oracle_68135361183973
// MI455X (gfx1250) — compile-verified
//
#include <hip/hip_runtime.h>

// ---------------- problem constants ----------------
#define N_    8
#define C_    8
#define T_    600
#define F_    257
#define LOADC 7.0710678118654755e-4f   // 0.001 / sqrt(2)

// float strides inside one (N,2,C,T,F) tensor
#define ST_T   F_               // 257
#define ST_C   (T_ * F_)        // 154200
#define ST_RI  (C_ * T_ * F_)   // 1233600
#define ST_N   (2 * ST_RI)

// ---------------- kernel-1 tiling ----------------
#define FT     16               // frequency bins per block (one wave each)
#define TC     8                // time frames per staged chunk
#define NWAVE  16
#define BLK    (NWAVE * 32)
#define NT     (T_ / TC)        // 75 chunks

// LDS geometry (words). TDM pads each 16-DWORD f-row by 1 DWORD -> stride 17.
// 4D tile order (f,c,ri,t) => LDS rows ordered rc = ri*8+c inside t: row = t*16+rc.
#define ROWW   17
#define MATW   (TC * 16 * ROWW)         // 2176 words per matrix tile (V or Y)
#define BUFW   (2 * MATW)               // 4352 words per buffer (V then Y)

typedef float v2f __attribute__((ext_vector_type(2)));
typedef float v8f __attribute__((ext_vector_type(8)));
typedef unsigned int v4u __attribute__((ext_vector_type(4)));
typedef int v4i __attribute__((ext_vector_type(4)));
typedef int v8i __attribute__((ext_vector_type(8)));

struct cpx { float re, im; };
__device__ __forceinline__ cpx cmul(cpx a, cpx b) {
    cpx r; r.re = a.re * b.re - a.im * b.im; r.im = a.re * b.im + a.im * b.re; return r;
}
__device__ __forceinline__ cpx csub(cpx a, cpx b) {
    cpx r; r.re = a.re - b.re; r.im = a.im - b.im; return r;
}

__device__ __forceinline__ unsigned lds_byte_addr(const void* p) {
    // generic shared pointer = {aperture_hi32, lds_offset_lo32}
    return (unsigned)(unsigned long long)(uintptr_t)p;
}

__device__ __forceinline__ v4u make_g0(unsigned long long ga, unsigned lds_byte) {
    v4u g0;
    g0.x = 1u;                                            // count=1, user mode
    g0.y = lds_byte;                                      // LDS dest (bytes)
    g0.z = (unsigned)(ga & 0xffffffffull);                // global_addr lo
    g0.w = (unsigned)((ga >> 32) & 0x01ffffffull)         // global_addr hi (57b)
         | 0x80000000u;                                   // type = 2 ("image")
    return g0;
}

__device__ __forceinline__ void tdm_issue(v4u g0, v8i g1, v4i g2, v4i g3) {
#if defined(__clang_major__) && (__clang_major__ >= 23)
    v8i g4; g4[0]=0; g4[1]=0; g4[2]=0; g4[3]=0; g4[4]=0; g4[5]=0; g4[6]=0; g4[7]=0;
    __builtin_amdgcn_tensor_load_to_lds(g0, g1, g2, g3, g4, 0);
#else
    __builtin_amdgcn_tensor_load_to_lds(g0, g1, g2, g3, 0);
#endif
}

// ============================================================
// Kernel 1: TDM-staged covariance Grams via v_wmma_f32_16x16x4_f32,
// then per-(n,f) 8x8 complex solve -> conj(H).  One wave per f bin.
// One 4D TDM descriptor per matrix per chunk:
//   dim0 = f (16, contig), dim1 = c (8, stride ST_C),
//   dim2 = ri (2, stride ST_RI), dim3 = t (TC, stride ST_T).
// ============================================================
__global__ __launch_bounds__(BLK)
void mvdr_cov_solve(const float* __restrict__ mix,
                    const float* __restrict__ noi,
                    float* __restrict__ Hc /* (N, F, C, 2) floats */) {
    __shared__ float SMEM[2 * BUFW];   // double-buffered V|Y tiles; DUMP overlaid later

    const int nfb  = (F_ + FT - 1) / FT;          // 17
    const int n    = blockIdx.x / nfb;
    const int f0   = (blockIdx.x % nfb) * FT;
    const unsigned Frem = (unsigned)(F_ - f0);    // >= 1; OOB f -> zeros
    const int tid  = threadIdx.x;
    const int wave = tid >> 5;                    // == f offset in tile
    const int lane = tid & 31;

    // ---- loop-invariant descriptor groups (uniform flow -> SALU/SGPR) ----
    v8i g1;
    g1[0] = (int)((2u << 16)        // data_size = 4 bytes
                | (1u << 20)        // pad_enable
                | (3u << 22)        // pad_interval: 16 DWORDs
                | (0u << 25));      // pad_amount: 1 DWORD
    g1[1] = (int)(Frem << 16);                          // tensor_dim0 lo16
    g1[2] = (int)((Frem >> 16) | (8u << 16));           // dim0 hi | tensor_dim1=8
    g1[3] = (int)(16u << 16);                           // tile_dim0 = 16
    g1[4] = (int)(8u | (2u << 16));                     // tile_dim1=8(c), tile_dim2=2(ri)
    g1[5] = (int)(unsigned)ST_C;                        // dim0_stride lo32 (c)
    g1[6] = (int)(((unsigned)ST_RI & 0xffffu) << 16);   // dim0_str hi16 | dim1_str lo16 (ri)
    g1[7] = (int)((unsigned)ST_RI >> 16);               // dim1_stride hi32

    v4i g2;
    g2[0] = 2;                    // tensor_dim2 (ri)
    g2[1] = T_;                   // tensor_dim3 (t)
    g2[2] = ST_T;                 // dim2_stride lo32 (t)
    g2[3] = (int)(8u << 16);      // dim2_str hi16 | tile_dim3 = TC
    v4i g3; g3[0] = 0; g3[1] = 0; g3[2] = 0; g3[3] = 0;

    const unsigned ldsB0 = lds_byte_addr(&SMEM[0]);
    const unsigned ldsB1 = lds_byte_addr(&SMEM[BUFW]);

    const float* noiN = noi + (size_t)n * ST_N + f0;
    const float* mixN = mix + (size_t)n * ST_N + f0;
    unsigned long long gaV = (unsigned long long)(uintptr_t)noiN;
    unsigned long long gaY = (unsigned long long)(uintptr_t)mixN;
    const unsigned long long STEP = (unsigned long long)TC * ST_T * 4u;  // bytes/chunk

    v8f accV = {0.f, 0.f, 0.f, 0.f, 0.f, 0.f, 0.f, 0.f};
    v8f accY = accV;

    const int hl   = lane & 15;                   // rc = ri*8 + c
    const int tofs = (lane >> 4) * 2;             // half-wave K offset

    // ---- issue chunk 0 ----
    {
        v4u g0V = make_g0(gaV, ldsB0);
        v4u g0Y = make_g0(gaY, ldsB0 + MATW * 4u);
        if (wave == 0) { tdm_issue(g0V, g1, g2, g3); tdm_issue(g0Y, g1, g2, g3); }
    }

    for (int tc = 0; tc < NT; ++tc) {
        if (tc + 1 < NT) {
            gaV += STEP; gaY += STEP;
            const unsigned lb = ((tc + 1) & 1) ? ldsB1 : ldsB0;
            v4u g0V = make_g0(gaV, lb);
            v4u g0Y = make_g0(gaY, lb + MATW * 4u);
            if (wave == 0) {
                tdm_issue(g0V, g1, g2, g3);
                tdm_issue(g0Y, g1, g2, g3);
                __builtin_amdgcn_s_wait_tensorcnt(2);   // chunk tc complete
            }
        } else {
            if (wave == 0) __builtin_amdgcn_s_wait_tensorcnt(0);
        }
        __syncthreads();

        // ---- Gram update: D += X * X^T, X = [Re;Im] stacked 16 x 4 ----
        const float* B = &SMEM[(tc & 1) * BUFW];
        #pragma unroll
        for (int s = 0; s < TC / 4; ++s) {
            const int t0 = s * 4 + tofs;
            const int i0 = ((t0    ) * 16 + hl) * ROWW + wave;
            const int i1 = ((t0 + 1) * 16 + hl) * ROWW + wave;
            v2f av; av.x = B[i0];        av.y = B[i1];
            v2f ay; ay.x = B[MATW + i0]; ay.y = B[MATW + i1];
            accV = __builtin_amdgcn_wmma_f32_16x16x4_f32(
                       false, av, false, av, (short)0, accV, false, false);
            accY = __builtin_amdgcn_wmma_f32_16x16x4_f32(
                       false, ay, false, ay, (short)0, accY, false, false);
        }
        __syncthreads();
    }

    // ---- dump 16x16 accumulators into overlaid LDS scratch ----
    float* D = &SMEM[wave * 512];
    #pragma unroll
    for (int c = 0; c < 8; ++c) {
        D[      c * 32 + lane] = accV[c];
        D[256 + c * 32 + lane] = accY[c];
    }
    __syncthreads();

    // ---- per-wave tiny complex solve on lane 0 ----
    const int f = f0 + wave;
    if (lane == 0 && f < F_) {
        cpx A[8][8], Bm[8][8];
        const float invT = 1.0f / (float)T_;
        for (int c = 0; c < 8; ++c) {
            for (int d = 0; d < 8; ++d) {
                // phi[c][d] = (RR+II) + i(IR-RI)
                float rr = D[c * 32 + d];
                float ri = D[c * 32 + 8 + d];
                float ir = D[c * 32 + 16 + d];
                float ii = D[c * 32 + 24 + d];
                A[c][d].re = (rr + ii) * invT + (c == d ? LOADC : 0.f);
                A[c][d].im = (ir - ri) * invT + (c == d ? LOADC : 0.f);
                rr = D[256 + c * 32 + d];
                ri = D[256 + c * 32 + 8 + d];
                ir = D[256 + c * 32 + 16 + d];
                ii = D[256 + c * 32 + 24 + d];
                Bm[c][d].re = (rr + ii) * invT;
                Bm[c][d].im = (ir - ri) * invT;
            }
        }
        // Gauss-Jordan: Bm := A^{-1} Bm  (= G)
        for (int k = 0; k < 8; ++k) {
            cpx p = A[k][k];
            float inv = 1.0f / (p.re * p.re + p.im * p.im);
            cpx ip; ip.re = p.re * inv; ip.im = -p.im * inv;
            for (int j = 0; j < 8; ++j) {
                A[k][j]  = cmul(A[k][j], ip);
                Bm[k][j] = cmul(Bm[k][j], ip);
            }
            for (int i = 0; i < 8; ++i) {
                if (i == k) continue;
                cpx fk = A[i][k];
                for (int j = 0; j < 8; ++j) {
                    A[i][j]  = csub(A[i][j],  cmul(fk, A[k][j]));
                    Bm[i][j] = csub(Bm[i][j], cmul(fk, Bm[k][j]));
                }
            }
        }
        // l = trace(G) - C*(1+1j)
        cpx l; l.re = -(float)C_; l.im = -(float)C_;
        for (int c = 0; c < 8; ++c) { l.re += Bm[c][c].re; l.im += Bm[c][c].im; }
        float ld = 1.0f / (l.re * l.re + l.im * l.im);
        cpx il; il.re = l.re * ld; il.im = -l.im * ld;

        float* outw = Hc + ((size_t)(n * F_ + f) * 8) * 2;
        for (int c = 0; c < 8; ++c) {
            cpx g = Bm[c][0];
            if (c == 0) g.re -= 1.0f;          // (G - I) column ref_mic
            cpx h = cmul(g, il);               // H = Ge / l
            outw[c * 2 + 0] =  h.re;           // store conj(H)
            outw[c * 2 + 1] = -h.im;
        }
    }
}

// ============================================================
// Kernel 2: beamform  X(n,t,f) = sum_c conj(H) * y  (streaming)
// ============================================================
__global__ __launch_bounds__(256)
void mvdr_apply(const float* __restrict__ mix,
                const float* __restrict__ Hc,
                float* __restrict__ out /* (N,2,1,T,F) */) {
    int idx = blockIdx.x * 256 + threadIdx.x;
    if (idx >= N_ * T_ * F_) return;
    int f = idx % F_;
    int t = (idx / F_) % T_;
    int n = idx / (T_ * F_);

    const float4* h4 = (const float4*)(Hc + ((size_t)(n * F_ + f) * 8) * 2);
    float4 h01 = h4[0], h23 = h4[1], h45 = h4[2], h67 = h4[3];
    float hre[8] = {h01.x, h01.z, h23.x, h23.z, h45.x, h45.z, h67.x, h67.z};
    float him[8] = {h01.y, h01.w, h23.y, h23.w, h45.y, h45.w, h67.y, h67.w};

    const float* yr = mix + (size_t)n * ST_N + (size_t)t * ST_T + (size_t)f;

    float xr = 0.f, xi = 0.f;
    #pragma unroll
    for (int c = 0; c < 8; ++c) {
        float ar = yr[(size_t)c * ST_C];
        float ai = yr[(size_t)c * ST_C + ST_RI];
        xr += hre[c] * ar - him[c] * ai;
        xi += hre[c] * ai + him[c] * ar;
    }
    size_t ob = (size_t)n * (2 * ST_C) + (size_t)t * ST_T + (size_t)f;
    out[ob]        = xr;
    out[ob + ST_C] = xi;
}

// ============================================================
extern "C" void kernel_launch(void* const* d_in, const int* in_sizes, int n_in,
                              void* d_out, int out_size, void* d_ws, size_t ws_size,
                              hipStream_t stream) {
    (void)in_sizes; (void)n_in; (void)out_size; (void)ws_size;
    const float* mix = (const float*)d_in[0];   // mixture
    const float* noi = (const float*)d_in[2];   // noise (target unused)
    float* Hc = (float*)d_ws;                   // N*F*8*2 floats = 131,584 B

    const int nfb = (F_ + FT - 1) / FT;         // 17
    mvdr_cov_solve<<<dim3(N_ * nfb), dim3(BLK), 0, stream>>>(mix, noi, Hc);

    const int total = N_ * T_ * F_;
    mvdr_apply<<<dim3((total + 255) / 256), dim3(256), 0, stream>>>(
        mix, Hc, (float*)d_out);
}